// LIF_Activation_24163486007651
// MI455X (gfx1250) — compile-verified
//
#include <hip/hip_runtime.h>
#include <cstdint>

// LIF forward scan, fixed shape x[64,1000,512].
// Memory-bound: 262 MB traffic -> ~11.3 us floor at 23.3 TB/s.
// CDNA5 path: ASYNCcnt-tracked global->LDS ring (48 timesteps/lane) for deep prefetch.

#define LIF_T    1000
#define LIF_U    512
#define UG       (LIF_U / 32)        // 16 u-groups of 32 lanes
#define DEPTH    48                  // timesteps in flight per lane (ASYNCcnt <= 48 < 63)
#define CHUNK    8                   // timesteps consumed per wait
#define LSTRIDE  52                  // floats per lane in ring: 16B-aligned, <=2-way bank conflict
#define WPB      8                   // waves per block
#define NTHREADS (WPB * 32)
#define MAIN_END (LIF_T - DEPTH)     // 952: last refill loads t=999 exactly

typedef __attribute__((address_space(3))) float lds_f;

#define WAIT_ASYNC(n) asm volatile("s_wait_asynccnt " #n ::: "memory")
#define WAIT_DS0()    asm volatile("s_wait_dscnt 0" ::: "memory")

// 8 async refills: LDS addr in VGPR, global = SGPR-pair base + small per-lane voffset (GVS form).
__device__ __forceinline__ void async_refill8(unsigned ldsSlot0B, unsigned u4,
                                              const float* gchunk) {
#pragma unroll
  for (int i = 0; i < CHUNK; ++i) {
    asm volatile("global_load_async_to_lds_b32 %0, %1, %2"
                 :: "v"(ldsSlot0B + (unsigned)(i * 4)),
                    "v"(u4 + (unsigned)(i * (LIF_U * 4))),
                    "s"(gchunk)
                 : "memory");
  }
}

__device__ __forceinline__ void lds_read8(const float* rlane, unsigned sbW, float* xv) {
  const float4 q0 = *(const float4*)(rlane + sbW);
  const float4 q1 = *(const float4*)(rlane + sbW + 4);
  xv[0] = q0.x; xv[1] = q0.y; xv[2] = q0.z; xv[3] = q0.w;
  xv[4] = q1.x; xv[5] = q1.y; xv[6] = q1.z; xv[7] = q1.w;
}

__device__ __forceinline__ void lif_compute8(const float* xv, float& vm,
                                             float wi, float cc, float* ochunk) {
#pragma unroll
  for (int i = 0; i < CHUNK; ++i) {
    const float kept = (vm < 1.0f) ? vm : 0.0f;          // hard-reset gate (Vm < 1)
    vm = __builtin_fmaf(cc, kept, wi * xv[i]);           // (1-wl)*Vm*keep + wi*x
    vm = (vm > 0.0f) ? vm : 0.0f;                        // relu
    const float sp = (vm > 1.0f) ? 1.0f : 0.0f;          // theta(Vm - 1)
    __builtin_nontemporal_store(sp, ochunk + i * LIF_U);
  }
}

__global__ __launch_bounds__(NTHREADS, 1)
void lif_scan_kernel(const float* __restrict__ x,
                     const float* __restrict__ w_i,
                     const float* __restrict__ w_l,
                     float* __restrict__ spikes) {
  __shared__ __align__(16) float ring[NTHREADS * LSTRIDE];   // 52 KB

  const int lane = threadIdx.x & 31;
  // Force wave-uniformity so b / global bases stay in SGPRs (s_add per chunk).
  const int wv  = __builtin_amdgcn_readfirstlane((int)(threadIdx.x >> 5));
  const int gwv = (int)blockIdx.x * WPB + wv;          // uniform wave id
  const int b   = gwv >> 4;                            // / UG (16)
  const int u   = ((gwv & (UG - 1)) << 5) + lane;

  const float wi = w_i[u];
  const float cc = 1.0f - w_l[u];

  const float*   gbase  = x + (size_t)b * (LIF_T * LIF_U);          // uniform
  float*         ochunk = spikes + (size_t)b * (LIF_T * LIF_U) + u; // per-lane
  const unsigned u4     = (unsigned)u * 4u;                         // per-lane voffset

  const float* rlane = &ring[threadIdx.x * LSTRIDE];
  unsigned ldsLane;
  {
    lds_f* r3 = (lds_f*)ring;                          // generic -> LDS addrspace
    ldsLane = (unsigned)(unsigned long long)(r3 + threadIdx.x * LSTRIDE);
  }

  // ---- prologue: put DEPTH=48 timesteps in flight (6 x 8 issues)
  {
    const float* g = gbase;
    unsigned sbB = 0;
#pragma unroll
    for (int p = 0; p < DEPTH / CHUNK; ++p) {
      async_refill8(ldsLane + sbB, u4, g);
      g += CHUNK * LIF_U;
      sbB += CHUNK * 4;
    }
  }

  float vm = 0.0f;
  unsigned sbW = 0;                                    // ring slot base (words)
  const float* gchunk = gbase + DEPTH * LIF_U;         // uniform, s_add per chunk

  // ---- main: 119 chunks with refill (refill t = tc+48 .. tc+55 <= 999, no clamp)
  for (int tc = 0; tc < MAIN_END; tc += CHUNK) {
    WAIT_ASYNC(40);                                    // oldest 8 of 48 complete
    float xv[CHUNK];
    lds_read8(rlane, sbW, xv);
    WAIT_DS0();                                        // reads landed before overwrite
    async_refill8(ldsLane + sbW * 4, u4, gchunk);
    lif_compute8(xv, vm, wi, cc, ochunk);
    gchunk += CHUNK * LIF_U;
    ochunk += CHUNK * LIF_U;
    sbW += CHUNK; if (sbW == DEPTH) sbW = 0;
  }

  // ---- drain: 6 chunks, descending wait immediates (tc = 952..992)
#define LIF_DRAIN(NWAIT)                                             \
  {                                                                  \
    WAIT_ASYNC(NWAIT);                                               \
    float xv[CHUNK];                                                 \
    lds_read8(rlane, sbW, xv);                                       \
    lif_compute8(xv, vm, wi, cc, ochunk);                            \
    ochunk += CHUNK * LIF_U;                                         \
    sbW += CHUNK; if (sbW == DEPTH) sbW = 0;                         \
  }
  LIF_DRAIN(40)
  LIF_DRAIN(32)
  LIF_DRAIN(24)
  LIF_DRAIN(16)
  LIF_DRAIN(8)
  LIF_DRAIN(0)
#undef LIF_DRAIN
}

extern "C" void kernel_launch(void* const* d_in, const int* in_sizes, int n_in,
                              void* d_out, int out_size, void* d_ws, size_t ws_size,
                              hipStream_t stream) {
  (void)n_in; (void)out_size; (void)d_ws; (void)ws_size;
  const float* x  = (const float*)d_in[0];
  const float* wi = (const float*)d_in[1];
  const float* wl = (const float*)d_in[2];
  float* out = (float*)d_out;

  const int B = in_sizes[0] / (LIF_T * LIF_U);         // 64
  const int blocks = (B * UG) / WPB;                   // 128 blocks x 256 threads
  lif_scan_kernel<<<blocks, NTHREADS, 0, stream>>>(x, wi, wl, out);
}